// TreeModelNN_11673721110717
// MI455X (gfx1250) — compile-verified
//
#include <hip/hip_runtime.h>

// TreeModelNN on MI455X (gfx1250).
// Roofline: W is 281 MB fp32 -> HBM-bound (~12 us floor @ 23.3 TB/s); compute
// (~2.5 GFLOP after the level-decomposition trick) is negligible, so we keep
// full fp32 precision and use V_WMMA_F32_16X16X4_F32.
// Per 16-leaf tile: h_out = sum over (level,node) passes of A(masked rows) x W[node].

#define LL    1024
#define HH    128
#define CCC   10
#define NNODE 2047
#define DCOL  34314
#define SP1   16384   // b0
#define SP2   16512   // W1
#define SP3   32896   // b1
#define SP4   33024   // W2
#define SP5   34304   // b2

typedef __attribute__((ext_vector_type(2))) float v2f;
typedef __attribute__((ext_vector_type(8))) float v8f;

// ---------------- delta = sum(W0^2) + sum_{i>=1} l1[i]/mut[i] ----------------

__global__ void __launch_bounds__(256)
delta_rows_kernel(const float* __restrict__ W, const float* __restrict__ heights,
                  const int* __restrict__ parent, float* __restrict__ rows) {
  const int i = blockIdx.x;
  const int tid = threadIdx.x;
  const float* wr = W + (size_t)i * DCOL;
  float acc = 0.f;
  if (i == 0) {
    for (int j = tid; j < DCOL; j += 256) { float w = wr[j]; acc += w * w; }
  } else {
    for (int j = tid; j < DCOL; j += 256) acc += fabsf(wr[j]);
  }
  __shared__ float red[256];
  red[tid] = acc;
  __syncthreads();
  for (int off = 128; off > 0; off >>= 1) {
    if (tid < off) red[tid] += red[tid + off];
    __syncthreads();
  }
  if (tid == 0) {
    float v = red[0];
    if (i > 0) {
      float mut = fmaxf(fabsf(heights[i] - heights[parent[i]]), 1e-7f);
      v = v / mut;
    }
    rows[i] = v;
  }
}

__global__ void __launch_bounds__(256)
delta_reduce_kernel(const float* __restrict__ rows, float* __restrict__ outp) {
  const int tid = threadIdx.x;
  float acc = 0.f;
  for (int j = tid; j < NNODE; j += 256) acc += rows[j];   // fixed order -> deterministic
  __shared__ float red[256];
  red[tid] = acc;
  __syncthreads();
  for (int off = 128; off > 0; off >>= 1) {
    if (tid < off) red[tid] += red[tid + off];
    __syncthreads();
  }
  if (tid == 0) outp[0] = red[0];
}

// ---------------- 128->128 layer: out = relu(in @ Weff + beff) ----------------
// grid (64 tiles, 2 col halves), 128 threads = 4 waves, each wave owns 16 cols.

__global__ void __launch_bounds__(128)
stage_gemm_kernel(const float* __restrict__ in, const float* __restrict__ W,
                  float* __restrict__ outp, int wOff, int bOff) {
  __shared__ float lds_h[16 * 132];   // padded stride: kills LDS bank conflicts
  __shared__ float lds_b[16 * 68];
  const int tile   = blockIdx.x;
  const int colGrp = blockIdx.y;                 // cols [64*g, 64*g+64)
  const int tid    = threadIdx.x;
  const int lane   = tid & 31;
  const int wv     = tid >> 5;
  const int m      = lane & 15;                  // A row (leaf row in tile)
  const int khalf  = lane >> 4;
  const int nLoc   = lane & 15;
  const int colBase = colGrp * 64 + wv * 16;     // this wave's 16 output cols

  // input tile 16x128 -> LDS
  for (int idx = tid; idx < 16 * 128; idx += 128) {
    int r = idx >> 7, c = idx & 127;
    lds_h[r * 132 + c] = in[(size_t)(tile * 16 + r) * 128 + c];
  }
  // effective bias for this block's 64 cols: sum over 11 ancestors
  for (int idx = tid; idx < 16 * 64; idx += 128) {
    int r = idx >> 6, c = idx & 63;
    int leaf = tile * 16 + r;
    float s = 0.f;
    for (int d = 0; d < 11; ++d) {
      int a = ((1024 + leaf) >> (10 - d)) - 1;
      s += W[(size_t)a * DCOL + bOff + colGrp * 64 + c];
    }
    lds_b[r * 68 + c] = s;
  }
  __syncthreads();

  v8f acc = {};
  for (int d = 0; d < 11; ++d) {
    const int shift = 10 - d;
    const int nb = (d < 7) ? 1 : (1 << (d - 6));
    const int rowsPer = 16 / nb;
    for (int s = 0; s < nb; ++s) {
      const int rlo = s * rowsPer;
      const int rhi = rlo + rowsPer;
      const int node = ((1024 + tile * 16 + rlo) >> shift) - 1;
      const float* __restrict__ Wb = W + (size_t)node * DCOL + wOff;
      const bool act = (m >= rlo) && (m < rhi);  // zero A rows outside node's leaves
      for (int k0 = 0; k0 < 32; ++k0) {
        const int kb = k0 * 4 + khalf * 2;
        v2f a = *(const v2f*)(lds_h + m * 132 + kb);
        if (!act) { a.x = 0.f; a.y = 0.f; }
        const float* bp = Wb + (size_t)kb * 128 + colBase + nLoc;
        v2f b; b.x = bp[0]; b.y = bp[128];
        acc = __builtin_amdgcn_wmma_f32_16x16x4_f32(false, a, false, b,
                                                    (short)0, acc, false, false);
      }
    }
  }

  // writeback: bias + relu
  for (int v = 0; v < 8; ++v) {
    const int mo = v + khalf * 8;
    const int leaf = tile * 16 + mo;
    float r = acc[v] + lds_b[mo * 68 + wv * 16 + nLoc];
    outp[(size_t)leaf * 128 + colBase + nLoc] = fmaxf(r, 0.f);
  }
}

// ---------------- 128->10 layer + softmax ----------------
// grid 64 tiles, 4 waves; passes split across waves, partials reduced via LDS.

__global__ void __launch_bounds__(128)
stage_out_kernel(const float* __restrict__ in, const float* __restrict__ W,
                 float* __restrict__ outp) {
  __shared__ float lds_h[16 * 132];
  __shared__ float part[4 * 16 * 16];
  const int tile  = blockIdx.x;
  const int tid   = threadIdx.x;
  const int lane  = tid & 31;
  const int wv    = tid >> 5;
  const int m     = lane & 15;
  const int khalf = lane >> 4;
  const int nLoc  = lane & 15;

  for (int idx = tid; idx < 16 * 128; idx += 128) {
    int r = idx >> 7, c = idx & 127;
    lds_h[r * 132 + c] = in[(size_t)(tile * 16 + r) * 128 + c];
  }
  __syncthreads();

  v8f acc = {};
  const bool ncol = (nLoc < CCC);
  int p = 0;
  for (int d = 0; d < 11; ++d) {
    const int shift = 10 - d;
    const int nb = (d < 7) ? 1 : (1 << (d - 6));
    const int rowsPer = 16 / nb;
    for (int s = 0; s < nb; ++s, ++p) {
      if ((p & 3) != wv) continue;               // wave-uniform: EXEC stays full
      const int rlo = s * rowsPer;
      const int rhi = rlo + rowsPer;
      const int node = ((1024 + tile * 16 + rlo) >> shift) - 1;
      const float* __restrict__ Wb = W + (size_t)node * DCOL + SP4;
      const bool act = (m >= rlo) && (m < rhi);
      for (int k0 = 0; k0 < 32; ++k0) {
        const int kb = k0 * 4 + khalf * 2;
        v2f a = *(const v2f*)(lds_h + m * 132 + kb);
        if (!act) { a.x = 0.f; a.y = 0.f; }
        v2f b;
        b.x = ncol ? Wb[(size_t)kb * CCC + nLoc]       : 0.f;
        b.y = ncol ? Wb[(size_t)(kb + 1) * CCC + nLoc] : 0.f;
        acc = __builtin_amdgcn_wmma_f32_16x16x4_f32(false, a, false, b,
                                                    (short)0, acc, false, false);
      }
    }
  }
  for (int v = 0; v < 8; ++v)
    part[wv * 256 + (v + khalf * 8) * 16 + nLoc] = acc[v];
  __syncthreads();

  if (tid < 16) {
    const int r = tid;
    const int leaf = tile * 16 + r;
    float logits[CCC];
    float mx = -3.4e38f;
    for (int c = 0; c < CCC; ++c) {
      float s = part[0 * 256 + r * 16 + c] + part[1 * 256 + r * 16 + c]
              + part[2 * 256 + r * 16 + c] + part[3 * 256 + r * 16 + c];
      for (int d = 0; d < 11; ++d) {
        int a = ((1024 + leaf) >> (10 - d)) - 1;
        s += W[(size_t)a * DCOL + SP5 + c];      // effective b2
      }
      logits[c] = s;
      mx = fmaxf(mx, s);
    }
    float se = 0.f;
    for (int c = 0; c < CCC; ++c) { logits[c] = __expf(logits[c] - mx); se += logits[c]; }
    float inv = 1.0f / se;
    for (int c = 0; c < CCC; ++c)
      outp[(size_t)leaf * CCC + c] = logits[c] * inv;
  }
}

// ---------------- launch ----------------

extern "C" void kernel_launch(void* const* d_in, const int* in_sizes, int n_in,
                              void* d_out, int out_size, void* d_ws, size_t ws_size,
                              hipStream_t stream) {
  (void)in_sizes; (void)n_in; (void)out_size; (void)ws_size;
  const float* x       = (const float*)d_in[0];   // (1024,128)
  const float* W       = (const float*)d_in[1];   // (2047,34314)
  const float* heights = (const float*)d_in[2];   // (2047,)
  const int*   parent  = (const int*)d_in[3];     // (2047,)
  float* out = (float*)d_out;                      // 10240 probs + 1 delta

  float* rows = (float*)d_ws;                      // 2048 floats
  float* h0   = rows + 2048;                       // 1024*128
  float* h1   = h0 + (size_t)LL * HH;              // 1024*128

  delta_rows_kernel<<<NNODE, 256, 0, stream>>>(W, heights, parent, rows);
  delta_reduce_kernel<<<1, 256, 0, stream>>>(rows, out + (size_t)LL * CCC);

  dim3 g(64, 2);
  stage_gemm_kernel<<<g, 128, 0, stream>>>(x,  W, h0, 0,   SP1);
  stage_gemm_kernel<<<g, 128, 0, stream>>>(h0, W, h1, SP2, SP3);
  stage_out_kernel<<<64, 128, 0, stream>>>(h1, W, out);
}